// FactorGraphMsgPassingLayer_NoDoubleCounting_12670153523304
// MI455X (gfx1250) — compile-verified
//
#include <hip/hip_runtime.h>
#include <math.h>

// ---------------------------------------------------------------------------
// Factor-graph BP iteration for MI455X (gfx1250, wave32, WMMA + async-LDS).
//
// Outputs (concatenated in d_out, all f32):
//   [0, V*2)                       var_beliefs'
//   [V*2, V*2+F*32)                factor_beliefs'
//   [.., +E*2)                     factorToVar messages
//   [.., +E*2)                     varToFactor messages
//
// Scratch lives inside the fb output region (fully rewritten by kernel E):
//   lse_per_axis[f][0..9]  at fb + f*32 + 0
//   scatter acc  [f][0..9] at fb + f*32 + 16
// ---------------------------------------------------------------------------

typedef __attribute__((ext_vector_type(2))) float v2f;
typedef __attribute__((ext_vector_type(8))) float v8f;

#define NSTATES 32
#define NK 10            // 5 axes * 2 states

__device__ __forceinline__ float maskneg(float x) {
    return (x == -INFINITY) ? 0.0f : x;
}

// ------------------------------ zero scratch -------------------------------
__global__ void k_zero(float* __restrict__ p, long n) {
    long i = (long)blockIdx.x * blockDim.x + threadIdx.x;
    long stride = (long)gridDim.x * blockDim.x;
    for (; i < n; i += stride) p[i] = 0.0f;
}

// ---------------- kernel A: per-factor axis marginals (WMMA) ---------------
// lse[f][d*2+x] = gmax_f + log( sum_{i: bit_{4-d}(i)=x} exp(b_i - gmax_f) )
// = row (d*2+x) of  M[16x32] x exp(B)[32x16]  via 8 chained WMMA_F32_16X16X4.
// Tile staging global->LDS uses GLOBAL_LOAD_ASYNC_TO_LDS_B128 (ASYNCcnt).
__global__ void k_factor_marg(const float* __restrict__ fbIn,
                              float* __restrict__ outFB, int nF) {
    __shared__ float raw[8][NSTATES * 16];   // per-wave [factor][state], 2KB
    __shared__ float tr [8][NSTATES * 16];   // per-wave [state][factor], 2KB
    int wave = threadIdx.x >> 5;
    int lane = threadIdx.x & 31;
    int waveGlobal = blockIdx.x * 8 + wave;
    int fbase = waveGlobal * 16;
    if (fbase + 16 > nF) fbase = nF - 16;    // clamp (duplicates write same data)

    int col = lane & 15;                     // factor within tile / A row
    int h   = lane >> 4;                     // half: states 0-15 vs 16-31

    // --- async stage 16 factors x 32 f32 (2KB) into LDS: 4 x 16B per lane ---
    {
        unsigned long long gbase =
            (unsigned long long)(fbIn + (size_t)fbase * NSTATES) + (unsigned)lane * 16u;
        unsigned lbase = (unsigned)(size_t)(&raw[wave][0]) + (unsigned)lane * 16u;
#pragma unroll
        for (int i = 0; i < 4; ++i) {
            unsigned long long ga = gbase + (unsigned)(i * 512);
            unsigned la = lbase + (unsigned)(i * 512);
            asm volatile("global_load_async_to_lds_b128 %0, %1, off"
                         :: "v"(la), "v"(ga) : "memory");
        }
        asm volatile("s_wait_asynccnt 0x0" ::: "memory");
    }

    // --- per-factor max + exp, transpose into tr ---
    float vals[16];
    const float4* rsrc = (const float4*)&raw[wave][col * NSTATES + h * 16];
#pragma unroll
    for (int j = 0; j < 4; ++j) {
        float4 t = rsrc[j];
        vals[4*j+0] = t.x; vals[4*j+1] = t.y; vals[4*j+2] = t.z; vals[4*j+3] = t.w;
    }
    float m = vals[0];
#pragma unroll
    for (int j = 1; j < 16; ++j) m = fmaxf(m, vals[j]);
    float gmax = fmaxf(m, __shfl_xor(m, 16));      // per-factor global max

#pragma unroll
    for (int j = 0; j < 16; ++j)
        tr[wave][(h * 16 + j) * 16 + col] = __expf(vals[j] - gmax);
    __syncthreads();

    // --- selection-matrix row as a 32-bit mask over states k ---
    // row r = d*2+x selects states with bit (4-d) == x
    unsigned rm = 0u;
    if (col < NK) {
        const unsigned C[5] = {0xFFFF0000u, 0xFF00FF00u, 0xF0F0F0F0u,
                               0xCCCCCCCCu, 0xAAAAAAAAu};
        unsigned cm = C[col >> 1];
        rm = (col & 1) ? cm : ~cm;
    }

    v8f acc = {};
#pragma unroll
    for (int kb = 0; kb < 8; ++kb) {
        int k0 = kb * 4 + h * 2;
        v2f a, b;
        a.x = (float)((rm >> k0) & 1u);
        a.y = (float)((rm >> (k0 + 1)) & 1u);
        b.x = tr[wave][(k0    ) * 16 + col];
        b.y = tr[wave][(k0 + 1) * 16 + col];
        acc = __builtin_amdgcn_wmma_f32_16x16x4_f32(
            false, a, false, b, (short)0, acc, false, false);
    }

    // D[m][n]: rows m = r + 8*h live in acc[r], column n = col (factor).
    float* dst = outFB + (size_t)(fbase + col) * NSTATES + h * 8;
#pragma unroll
    for (int r = 0; r < 8; ++r) {
        int mrow = h * 8 + r;
        if (mrow < NK) dst[r] = gmax + __logf(acc[r]);
    }
}

// ---------------- kernel B: edge pass 1 (factor -> var messages) -----------
__global__ void k_edge1(const float* __restrict__ fbRegion,
                        const float* __restrict__ prv_v2f,
                        const int* __restrict__ f_idx,
                        const int* __restrict__ v_idx,
                        const int* __restrict__ ev_idx,
                        float* __restrict__ f2v_out,
                        float* __restrict__ vb_acc, int nE) {
    int e = blockIdx.x * blockDim.x + threadIdx.x;
    if (e >= nE) return;
    int f = f_idx[e], d = ev_idx[e], v = v_idx[e];
    const float* l = fbRegion + (size_t)f * NSTATES + d * 2;   // lse stash
    float m0 = l[0] - maskneg(prv_v2f[2 * e]);
    float m1 = l[1] - maskneg(prv_v2f[2 * e + 1]);
    f2v_out[2 * e]     = m0;
    f2v_out[2 * e + 1] = m1;
    atomicAdd(vb_acc + 2 * v,     m0);
    atomicAdd(vb_acc + 2 * v + 1, m1);
}

// ---------------- kernel C: normalize var beliefs --------------------------
__global__ void k_norm_vb(float* __restrict__ vb, int nV) {
    int v = blockIdx.x * blockDim.x + threadIdx.x;
    if (v >= nV) return;
    float b0 = vb[2 * v], b1 = vb[2 * v + 1];
    float m = fmaxf(b0, b1);
    float l = m + __logf(__expf(b0 - m) + __expf(b1 - m));
    vb[2 * v]     = b0 - l;
    vb[2 * v + 1] = b1 - l;
}

// ---------------- kernel D: edge pass 2 (var -> factor messages) -----------
__global__ void k_edge2(const float* __restrict__ vb,
                        const float* __restrict__ prv_f2v,
                        const int* __restrict__ f_idx,
                        const int* __restrict__ v_idx,
                        const int* __restrict__ ev_idx,
                        float* __restrict__ v2f_out,
                        float* __restrict__ fbRegion, int nE) {
    int e = blockIdx.x * blockDim.x + threadIdx.x;
    if (e >= nE) return;
    int f = f_idx[e], d = ev_idx[e], v = v_idx[e];
    float u0 = vb[2 * v]     - maskneg(prv_f2v[2 * e]);
    float u1 = vb[2 * v + 1] - maskneg(prv_f2v[2 * e + 1]);
    v2f_out[2 * e]     = u0;
    v2f_out[2 * e + 1] = u1;
    float* acc = fbRegion + (size_t)f * NSTATES + 16 + d * 2;  // scatter acc
    atomicAdd(acc,     u0);
    atomicAdd(acc + 1, u1);
}

// ---------------- kernel E: expand + potentials + normalize (WMMA) ---------
// fb[f][i] = sum_k MT[i][k] * acc[f][k] + pot[f][i], then LSE-normalize over 32.
// MT is 32x10 (K padded to 12): two 16-row A tiles, 3 k-steps -> 6 WMMAs/tile.
__global__ void k_factor_out(const float* __restrict__ pot,
                             float* __restrict__ outFB, int nF) {
    int lane = threadIdx.x & 31;
    int waveGlobal = blockIdx.x * (blockDim.x >> 5) + (threadIdx.x >> 5);
    int fbase = waveGlobal * 16;
    if (fbase >= nF) return;

    int col = lane & 15;
    int h   = lane >> 4;
    size_t f = (size_t)fbase + col;
    const float* accp = outFB + f * NSTATES + 16;

    v2f b[3];
#pragma unroll
    for (int kb = 0; kb < 3; ++kb) {
        int k0 = kb * 4 + h * 2;
        if (k0 < NK) {                       // k0 even; k0+1 <= 9 when k0 < 10
            b[kb] = *(const v2f*)(accp + k0);
        } else {
            b[kb].x = 0.0f; b[kb].y = 0.0f;
        }
    }

    // MT row masks over k = d*2 + x: bit set iff bit_{4-d}(state) == x
    unsigned cm0 = 0u, cm1 = 0u;
#pragma unroll
    for (int d = 0; d < 5; ++d) {
        cm0 |= 1u << (2 * d + ((col >> (4 - d)) & 1));
        cm1 |= 1u << (2 * d + (((col + 16) >> (4 - d)) & 1));
    }

    v8f d0 = {}, d1 = {};
#pragma unroll
    for (int kb = 0; kb < 3; ++kb) {
        int k0 = kb * 4 + h * 2;
        v2f a0, a1;
        a0.x = (float)((cm0 >> k0) & 1u);
        a0.y = (float)((cm0 >> (k0 + 1)) & 1u);
        a1.x = (float)((cm1 >> k0) & 1u);
        a1.y = (float)((cm1 >> (k0 + 1)) & 1u);
        d0 = __builtin_amdgcn_wmma_f32_16x16x4_f32(
            false, a0, false, b[kb], (short)0, d0, false, false);
        d1 = __builtin_amdgcn_wmma_f32_16x16x4_f32(
            false, a1, false, b[kb], (short)0, d1, false, false);
    }

    // add potentials: lane holds states (h*8 + r) and (16 + h*8 + r) of factor f
    const float* pp = pot + f * NSTATES;
    float t0[8], t1[8];
#pragma unroll
    for (int r = 0; r < 8; ++r) {
        t0[r] = d0[r] + pp[h * 8 + r];
        t1[r] = d1[r] + pp[16 + h * 8 + r];
    }
    float m = t0[0];
#pragma unroll
    for (int r = 1; r < 8; ++r) m = fmaxf(m, t0[r]);
#pragma unroll
    for (int r = 0; r < 8; ++r) m = fmaxf(m, t1[r]);
    float gm = fmaxf(m, __shfl_xor(m, 16));
    float s = 0.0f;
#pragma unroll
    for (int r = 0; r < 8; ++r) s += __expf(t0[r] - gm) + __expf(t1[r] - gm);
    float gs = s + __shfl_xor(s, 16);
    float l = gm + __logf(gs);

    float* dst = outFB + f * NSTATES;
#pragma unroll
    for (int r = 0; r < 8; ++r) {
        dst[h * 8 + r]      = t0[r] - l;
        dst[16 + h * 8 + r] = t1[r] - l;
    }
}

// ---------------------------------------------------------------------------
extern "C" void kernel_launch(void* const* d_in, const int* in_sizes, int n_in,
                              void* d_out, int out_size, void* d_ws, size_t ws_size,
                              hipStream_t stream) {
    const float* fb_in    = (const float*)d_in[0];   // [F, 32]
    // d_in[1] var_beliefs: only used for shape in the reference
    const float* pot      = (const float*)d_in[2];   // [F, 32]
    const float* prv_v2f  = (const float*)d_in[3];   // [E, 2]
    const float* prv_f2v  = (const float*)d_in[4];   // [E, 2]
    const int*   f_idx    = (const int*)d_in[5];     // [E]
    const int*   v_idx    = (const int*)d_in[6];     // [E]
    const int*   ev_idx   = (const int*)d_in[7];     // [E]

    const int F_ = in_sizes[0] / NSTATES;
    const int V_ = in_sizes[1] / 2;
    const int E_ = in_sizes[5];

    float* out     = (float*)d_out;
    float* out_vb  = out;                               // [V, 2]
    float* out_fb  = out + (size_t)V_ * 2;              // [F, 32] (+ scratch)
    float* out_f2v = out_fb + (size_t)F_ * NSTATES;     // [E, 2]
    float* out_v2f = out_f2v + (size_t)E_ * 2;          // [E, 2]

    // 0) zero var accumulator + fb region (holds lse stash & scatter acc)
    long nzero = (long)V_ * 2 + (long)F_ * NSTATES;
    k_zero<<<2048, 256, 0, stream>>>(out, nzero);

    // 1) per-factor axis marginals via async-LDS staging + WMMA
    int wavesA = (F_ + 15) / 16;
    int blocksA = (wavesA + 7) / 8;
    k_factor_marg<<<blocksA, 256, 0, stream>>>(fb_in, out_fb, F_);

    // 2) edge pass 1: f2v messages + var-belief scatter
    k_edge1<<<(E_ + 255) / 256, 256, 0, stream>>>(
        out_fb, prv_v2f, f_idx, v_idx, ev_idx, out_f2v, out_vb, E_);

    // 3) normalize var beliefs
    k_norm_vb<<<(V_ + 255) / 256, 256, 0, stream>>>(out_vb, V_);

    // 4) edge pass 2: v2f messages + factor-accumulator scatter
    k_edge2<<<(E_ + 255) / 256, 256, 0, stream>>>(
        out_vb, prv_f2v, f_idx, v_idx, ev_idx, out_v2f, out_fb, E_);

    // 5) expand accumulator + potentials, LSE-normalize factors via WMMA
    k_factor_out<<<blocksA, 256, 0, stream>>>(pot, out_fb, F_);
}